// Quadratic_77936476553489
// MI455X (gfx1250) — compile-verified
//
#include <hip/hip_runtime.h>

typedef __attribute__((ext_vector_type(2))) float v2f;
typedef __attribute__((ext_vector_type(8))) float v8f;

#define DD 11   // feature dim

// DPP-based shuffle-add: x + lanepermute(x).  quad_perm / row_ror only
// (baseline DPP16 controls), so the whole reduction stays on the VALU —
// no LDS (ds_bpermute) traffic.
#define DPP_XOR1 0xB1   // quad_perm(1,0,3,2)
#define DPP_XOR2 0x4E   // quad_perm(2,3,0,1)
#define DPP_ROR4 0x124  // row_ror:4  (preserves lane bits 0-1)
#define DPP_ROR8 0x128  // row_ror:8  (preserves lane bits 0-2)

template <int CTRL>
__device__ __forceinline__ float dpp_xadd(float x) {
    int r = __builtin_amdgcn_update_dpp(0, __float_as_int(x), CTRL, 0xF, 0xF, true);
    return x + __int_as_float(r);
}

// out[n] = b + W.x_n + x_n^T Q x_n  ==  x~_n^T M x~_n,
// x~ = [x;1] (12), M = [[Q(11x11), 0],[W(1x11), b]] (12x12, zero-padded to 12x16).
// Per 16-row tile: P = X~ @ M via 3x v_wmma_f32_16x16x4_f32, then
// out[m] = sum_c P[m,c] * X~[m,c] via elementwise mul + packed DPP reduction.
__global__ __launch_bounds__(256) void quad_wmma_kernel(
    const float* __restrict__ X, const float* __restrict__ bp,
    const float* __restrict__ W, const float* __restrict__ Q,
    float* __restrict__ out, int n_tiles)
{
    const int lane = threadIdx.x & 31;
    const int half = lane >> 4;     // 0: lanes 0-15, 1: lanes 16-31
    const int lcol = lane & 15;

    // ---- Constant B-fragments of M (3 K-slices of 4 rows each), built once ----
    // B layout (4x16 f32): VGPR0 = row k0 (lanes0-15) / row k0+2 (lanes16-31),
    //                      VGPR1 = row k0+1 / k0+3.  Per lane: rows r0, r0+1.
    v2f bf[3];
#pragma unroll
    for (int s = 0; s < 3; ++s) {
        const int r0 = 4 * s + 2 * half;        // r0 in {0,2,4,6,8,10}
        float m0 = 0.0f, m1 = 0.0f;
        if (lcol < DD) {
            m0 = Q[r0 * DD + lcol];                                   // r0 <= 10
            m1 = (r0 + 1 < DD) ? Q[(r0 + 1) * DD + lcol] : W[lcol];   // row 11 = W
        } else if (lcol == DD) {
            m1 = (r0 + 1 == DD) ? bp[0] : 0.0f;                       // M[11][11] = b
        } // cols 12..15 of M are zero padding
        bf[s].x = m0; bf[s].y = m1;
    }

    // Per-lane constants for the C-layout X~ blend: v = raw*xm + xa
    const int   cc = (lcol < DD) ? lcol : (DD - 1);   // clamped, always in-bounds
    const float xm = (lcol < DD)  ? 1.0f : 0.0f;
    const float xa = (lcol == DD) ? 1.0f : 0.0f;

    const int waves_per_blk = blockDim.x >> 5;
    const int wid     = blockIdx.x * waves_per_blk + (threadIdx.x >> 5);
    const int wstride = gridDim.x * waves_per_blk;

    for (int tile = wid; tile < n_tiles; tile += wstride) {
        const int row0 = tile << 4;

        // Prefetch the next grid-stride tile's 704-B window (global_prefetch_b8).
        {
            const int nt = tile + wstride;
            if (nt < n_tiles) {
                const float* p = X + (size_t)(nt << 4) * DD + (size_t)(lane & 7) * 22;
                __builtin_prefetch(p, 0, 1);
            }
        }

        // ---- A-fragments (X~ rows, K-slices of 4 cols), branchless ----
        // A layout (16x4 f32): lane row = lcol; VGPR0 = col k0/k0+2, VGPR1 = k0+1/k0+3.
        const float* xr = X + (size_t)(row0 + lcol) * DD;
        const float* p2 = xr + 2 * half;          // per-lane col base
        v2f af[3];
        af[0].x = p2[0]; af[0].y = p2[1];
        af[1].x = p2[4]; af[1].y = p2[5];
        af[2].x = p2[8];                          // col 8+2h  (<= 10, in-bounds)
        {
            const float r9 = xr[9];               // col 9, always in-bounds
            af[2].y = half ? 1.0f : r9;           // logical col 11 -> x~ = 1
        }

        // ---- X~ in C/D layout: VGPR r = row (row0+r+8*half), col lcol ----
        float xc[8];
        {
            const float* xb = X + (size_t)(row0 + (half << 3)) * DD + cc;
#pragma unroll
            for (int r = 0; r < 8; ++r)
                xc[r] = fmaf(xb[r * DD], xm, xa); // raw for col<11, 1 at col 11, else 0
        }

        // ---- P = X~ @ M : three chained f32 WMMAs (K = 12) ----
        v8f acc = {};
        acc = __builtin_amdgcn_wmma_f32_16x16x4_f32(false, af[0], false, bf[0],
                                                    (short)0, acc, false, false);
        acc = __builtin_amdgcn_wmma_f32_16x16x4_f32(false, af[1], false, bf[1],
                                                    (short)0, acc, false, false);
        acc = __builtin_amdgcn_wmma_f32_16x16x4_f32(false, af[2], false, bf[2],
                                                    (short)0, acc, false, false);

        // ---- out[m] = sum_c P[m,c]*X~[m,c]: packed DPP reduction ----
        float t[8];
#pragma unroll
        for (int r = 0; r < 8; ++r) {
            float v = acc[r] * xc[r];
            t[r] = dpp_xadd<DPP_XOR1>(v);         // pairs done; dup mod 2
        }
        const bool b0 = (lane & 1) != 0;
        float q0 = b0 ? t[4] : t[0];
        float q1 = b0 ? t[5] : t[1];
        float q2 = b0 ? t[6] : t[2];
        float q3 = b0 ? t[7] : t[3];
        q0 = dpp_xadd<DPP_XOR2>(q0);              // quads done; dup mod 4
        q1 = dpp_xadd<DPP_XOR2>(q1);
        q2 = dpp_xadd<DPP_XOR2>(q2);
        q3 = dpp_xadd<DPP_XOR2>(q3);
        const bool b1 = (lane & 2) != 0;
        float w0 = b1 ? q2 : q0;                  // row r* = 4*b0 + 2*b1 + j
        float w1 = b1 ? q3 : q1;
        w0 = dpp_xadd<DPP_ROR4>(w0);              // rotations preserve bits 0-1
        w1 = dpp_xadd<DPP_ROR4>(w1);
        w0 = dpp_xadd<DPP_ROR8>(w0);              // full 16-lane sums
        w1 = dpp_xadd<DPP_ROR8>(w1);

        // lanes 0-3 / 16-19 each hold two consecutive row results -> b64 store
        if (lcol < 4) {
            const int rbase = row0 + (half << 3) + ((lcol & 1) << 2) + (lcol & 2);
            v2f o; o.x = w0; o.y = w1;
            *(v2f*)(out + rbase) = o;
        }
    }
}

// Scalar tail for N % 16 rows (N = 4,000,000 is divisible; defensive only).
__global__ void quad_tail_kernel(const float* __restrict__ X,
                                 const float* __restrict__ bp,
                                 const float* __restrict__ W,
                                 const float* __restrict__ Q,
                                 float* __restrict__ out, int start, int ntot)
{
    int i = start + blockIdx.x * blockDim.x + threadIdx.x;
    if (i >= ntot) return;
    const float* x = X + (size_t)i * DD;
    float lin = 0.0f, quad = 0.0f;
#pragma unroll
    for (int r = 0; r < DD; ++r) {
        lin = fmaf(W[r], x[r], lin);
        float yr = 0.0f;
#pragma unroll
        for (int c = 0; c < DD; ++c) yr = fmaf(Q[r * DD + c], x[c], yr);
        quad = fmaf(x[r], yr, quad);
    }
    out[i] = bp[0] + lin + quad;
}

extern "C" void kernel_launch(void* const* d_in, const int* in_sizes, int n_in,
                              void* d_out, int out_size, void* d_ws, size_t ws_size,
                              hipStream_t stream)
{
    const float* X  = (const float*)d_in[0];
    const float* b  = (const float*)d_in[1];
    const float* W  = (const float*)d_in[2];
    const float* Q  = (const float*)d_in[3];
    float* out = (float*)d_out;

    const int ntot    = in_sizes[0] / DD;   // 4,000,000
    const int n_tiles = ntot / 16;          // 250,000
    const int rem_start = n_tiles * 16;

    if (n_tiles > 0) {
        int blocks = (n_tiles + 7) / 8;     // 8 waves per 256-thread block
        if (blocks > 4096) blocks = 4096;   // grid-stride covers the rest
        quad_wmma_kernel<<<blocks, 256, 0, stream>>>(X, b, W, Q, out, n_tiles);
    }
    const int rem = ntot - rem_start;
    if (rem > 0) {
        quad_tail_kernel<<<(rem + 255) / 256, 256, 0, stream>>>(X, b, W, Q, out,
                                                                rem_start, ntot);
    }
}